// PointcloudGrouping_62268435857483
// MI455X (gfx1250) — compile-verified
//
#include <hip/hip_runtime.h>
#include <hip/hip_bf16.h>

typedef float v2f __attribute__((ext_vector_type(2)));
typedef float v8f __attribute__((ext_vector_type(8)));

#define NPTS   16384
#define NBATCH 8
#define NGRP   512
#define KORIG  160      // 5 * GROUP_SIZE
#define GSIZE  32
#define RAD2   0.04f    // 0.2^2
#define GRAD   0.2f
#define GROUPS_ELEMS (NBATCH * NGRP * GSIZE * 4)   // 524288 floats, then centers

// ---------------------------------------------------------------------------
// Kernel 1: farthest point sampling. One 1024-thread (32-wave) block / batch.
// xyz + running min-distance live in registers (16 pts/thread); each of the
// 512 sequential steps is VALU + shfl/LDS argmax reduction only.
// ---------------------------------------------------------------------------
__global__ __launch_bounds__(1024)
void fps_kernel(const float4* __restrict__ pts4,
                const int*    __restrict__ lengths,
                float*        __restrict__ centers)
{
    const int b    = blockIdx.x;
    const int tid  = threadIdx.x;
    const int wave = tid >> 5;
    const int lane = tid & 31;
    const int len  = lengths[b];

    __shared__ float wval[32];
    __shared__ int   widx[32];
    __shared__ int   curIdx;

    float px[16], py[16], pz[16], md[16];
    #pragma unroll
    for (int t = 0; t < 16; ++t) {
        int i = t * 1024 + tid;
        float4 p = pts4[(size_t)b * NPTS + i];
        px[t] = p.x; py[t] = p.y; pz[t] = p.z;
        md[t] = (i < len) ? __builtin_inff() : -__builtin_inff();
    }
    if (tid == 0) curIdx = 0;
    __syncthreads();

    for (int k = 0; k < NGRP; ++k) {
        int cur = curIdx;
        float4 cp = pts4[(size_t)b * NPTS + cur];   // uniform broadcast load
        if (tid == 0) {
            centers[((size_t)b * NGRP + k) * 3 + 0] = cp.x;
            centers[((size_t)b * NGRP + k) * 3 + 1] = cp.y;
            centers[((size_t)b * NGRP + k) * 3 + 2] = cp.z;
        }
        // update min-dist, local argmax (tie -> smallest global index)
        float bv; int bidx;
        {
            float dx = px[0] - cp.x, dy = py[0] - cp.y, dz = pz[0] - cp.z;
            md[0] = fminf(md[0], dx*dx + dy*dy + dz*dz);
            bv = md[0]; bidx = tid;
        }
        #pragma unroll
        for (int t = 1; t < 16; ++t) {
            float dx = px[t] - cp.x, dy = py[t] - cp.y, dz = pz[t] - cp.z;
            md[t] = fminf(md[t], dx*dx + dy*dy + dz*dz);
            int i = t * 1024 + tid;
            if (md[t] > bv) { bv = md[t]; bidx = i; }
        }
        // wave32 butterfly reduction (max, min-index tiebreak)
        #pragma unroll
        for (int off = 16; off > 0; off >>= 1) {
            float ov = __shfl_xor(bv, off, 32);
            int   oi = __shfl_xor(bidx, off, 32);
            if (ov > bv || (ov == bv && oi < bidx)) { bv = ov; bidx = oi; }
        }
        if (lane == 0) { wval[wave] = bv; widx[wave] = bidx; }
        __syncthreads();
        if (tid < 32) {                 // wave 0 reduces the 32 wave results
            float v = wval[lane]; int i2 = widx[lane];
            #pragma unroll
            for (int off = 16; off > 0; off >>= 1) {
                float ov = __shfl_xor(v, off, 32);
                int   oi = __shfl_xor(i2, off, 32);
                if (ov > v || (ov == v && oi < i2)) { v = ov; i2 = oi; }
            }
            if (lane == 0) curIdx = i2;
        }
        __syncthreads();
    }
}

// ---------------------------------------------------------------------------
// Kernel 2: ball query (WMMA f32 16x16x4 distance tiles, ballot membership)
// + in-index-order candidate collection + top-32-by-energy + relative gather.
// One 256-thread (8-wave) block handles 16 groups of one batch.
// ---------------------------------------------------------------------------
__global__ __launch_bounds__(256)
void group_kernel(const float4* __restrict__ pts4,
                  const int*    __restrict__ lengths,
                  const float*  __restrict__ centers,
                  float4*       __restrict__ groups)
{
    const int blk   = blockIdx.x;
    const int b     = blk >> 5;          // 32 group-tiles per batch
    const int gbase = (blk & 31) * 16;
    const int tid   = threadIdx.x;
    const int wave  = tid >> 5;
    const int lane  = tid & 31;
    const int len   = lengths[b];
    const float NEGINF = -__builtin_inff();

    __shared__ float    cx[16], cy[16], cz[16], cn2[16];
    __shared__ unsigned tmask[8][16];          // [tile][group] membership bits
    __shared__ float    echunk[128];           // energies of current 128-pt chunk
    __shared__ int      cand[16][KORIG];
    __shared__ float    cande[16][KORIG];
    __shared__ int      ccount[16];

    if (tid < 16) {
        int g = gbase + tid;
        float x = centers[((size_t)b * NGRP + g) * 3 + 0];
        float y = centers[((size_t)b * NGRP + g) * 3 + 1];
        float z = centers[((size_t)b * NGRP + g) * 3 + 2];
        cx[tid] = x; cy[tid] = y; cz[tid] = z;
        cn2[tid] = x*x + y*y + z*z;
        ccount[tid] = 0;
    }
    __syncthreads();

    // A operand: 16x4 f32 (M=16 centers, K=xyz0).
    // ISA layout: lanes 0-15 hold K=0,1 ; lanes 16-31 hold K=2,3.
    v2f a;
    a.x = (lane < 16) ? cx[lane] : cz[lane - 16];
    a.y = (lane < 16) ? cy[lane] : 0.0f;

    // Hoist loop-invariant center norms into registers (this lane's 8 rows).
    float cn2r[8];
    #pragma unroll
    for (int r = 0; r < 8; ++r)
        cn2r[r] = cn2[(lane < 16) ? r : (r + 8)];

    for (int base = 0; base < NPTS; base += 128) {
        int q = base + wave * 16 + (lane & 15);
        float4 p = pts4[(size_t)b * NPTS + q];
        if (base + 128 < NPTS)
            __builtin_prefetch(&pts4[(size_t)b * NPTS + q + 128], 0, 3);
        float pn2  = p.x*p.x + p.y*p.y + p.z*p.z;
        bool valid = q < len;
        if (lane < 16) echunk[wave * 16 + lane] = p.w;
        __syncthreads();   // echunk ready; previous append phase complete

        // B operand: 4x16 f32 (K=xyz0, N=16 points), same half-wave K split.
        v2f bvec;
        bvec.x = (lane < 16) ? p.x : p.z;
        bvec.y = (lane < 16) ? p.y : 0.0f;
        v8f acc = {};
        acc = __builtin_amdgcn_wmma_f32_16x16x4_f32(
                  /*neg_a=*/false, a, /*neg_b=*/false, bvec,
                  /*c_mod=*/(short)0, acc, /*reuse_a=*/false, /*reuse_b=*/false);

        // d2 = |c|^2 + |p|^2 - 2 c.p ; wave ballot -> 16-bit masks for the
        // row pair (group r in bits [15:0], group r+8 in bits [31:16]).
        #pragma unroll
        for (int r = 0; r < 8; ++r) {
            float d2 = cn2r[r] + pn2 - 2.0f * acc[r];
            unsigned mask = __builtin_amdgcn_ballot_w32(valid && (d2 < RAD2));
            if (lane == 0) {
                tmask[wave][r]     = mask & 0xffffu;
                tmask[wave][r + 8] = mask >> 16;
            }
        }
        __syncthreads();   // tmask ready

        // append in ascending point-index order (== top_k of N-n score)
        int myfull = 1;
        if (tid < 16) {
            int g = tid;
            int cnt = ccount[g];
            for (int t = 0; t < 8 && cnt < KORIG; ++t) {
                unsigned m = tmask[t][g];
                while (m && cnt < KORIG) {
                    int n = __builtin_ctz(m);
                    m &= m - 1u;
                    int pos = t * 16 + n;
                    cand[g][cnt]  = base + pos;
                    cande[g][cnt] = echunk[pos];
                    ++cnt;
                }
            }
            ccount[g] = cnt;
            myfull = (cnt >= KORIG) ? 1 : 0;
        }
        if (__syncthreads_and(myfull)) break;   // fused barrier + all-reduce
    }

    // ---- top-32 by energy per group (stable: ties -> smaller position) ----
    #pragma unroll 1
    for (int gsel = 0; gsel < 2; ++gsel) {
        int g   = wave * 2 + gsel;           // local group handled by this wave
        int cnt = ccount[g];
        float ev[5];
        #pragma unroll
        for (int j = 0; j < 5; ++j) {
            int pos = lane + j * 32;
            ev[j] = (pos < cnt) ? cande[g][pos] : NEGINF;
        }
        int mysel = -1;                      // lane k keeps k-th selection
        for (int k = 0; k < GSIZE; ++k) {
            float bvv = NEGINF; int bpos = 0x7fffffff;
            #pragma unroll
            for (int j = 0; j < 5; ++j) {
                int pos = lane + j * 32;
                if (ev[j] > bvv || (ev[j] == bvv && pos < bpos)) { bvv = ev[j]; bpos = pos; }
            }
            #pragma unroll
            for (int off = 16; off > 0; off >>= 1) {
                float ov = __shfl_xor(bvv, off, 32);
                int   op = __shfl_xor(bpos, off, 32);
                if (ov > bvv || (ov == bvv && op < bpos)) { bvv = ov; bpos = op; }
            }
            int sel = (bvv > NEGINF) ? cand[g][bpos] : -1;
            if (bvv > NEGINF && lane == (bpos & 31)) {
                #pragma unroll
                for (int j = 0; j < 5; ++j)
                    if (j == (bpos >> 5)) ev[j] = NEGINF;   // mark used
            }
            if (lane == k) mysel = sel;
        }
        int first = __shfl(mysel, 0, 32);
        int idx   = (mysel < 0) ? first : mysel;

        float gx = cx[g], gy = cy[g], gz = cz[g];
        float4 o;
        if (idx >= 0) {
            float4 p = pts4[(size_t)b * NPTS + idx];
            o.x = (p.x - gx) / GRAD;
            o.y = (p.y - gy) / GRAD;
            o.z = (p.z - gz) / GRAD;
            o.w = p.w / GRAD;
        } else {                             // empty group: gather==0
            o.x = (0.0f - gx) / GRAD;
            o.y = (0.0f - gy) / GRAD;
            o.z = (0.0f - gz) / GRAD;
            o.w = 0.0f;
        }
        groups[((size_t)(b * NGRP + gbase + g)) * GSIZE + lane] = o;
    }
}

extern "C" void kernel_launch(void* const* d_in, const int* in_sizes, int n_in,
                              void* d_out, int out_size, void* d_ws, size_t ws_size,
                              hipStream_t stream)
{
    const float4* pts4    = (const float4*)d_in[0];
    const int*    lengths = (const int*)d_in[1];
    float*        outF    = (float*)d_out;
    float*        centers = outF + GROUPS_ELEMS;   // (B, 512, 3) tail of d_out

    fps_kernel<<<NBATCH, 1024, 0, stream>>>(pts4, lengths, centers);
    group_kernel<<<NBATCH * (NGRP / 16), 256, 0, stream>>>(
        pts4, lengths, centers, (float4*)d_out);
}